// FullOnlineHadamardHook_52510270161578
// MI455X (gfx1250) — compile-verified
//
#include <hip/hip_runtime.h>

// FullOnlineHadamardHook for MI455X (gfx1250, wave32, WMMA)
//
// out[b] = had_K(172x172) @ ( FWHT_64(X[b]) * 1/sqrt(11008) ),  b in [0, 8192)
//
// - FWHT_64 per row in registers (VALU, co-executes with matrix pipe)
// - 172x172 x 172x64 GEMM per batch on the matrix pipe with
//   V_WMMA_F32_16X16X4_F32 (full fp32 precision, matches reference dtype)
// - had_K cached in LDS once per workgroup (CDNA5 WGP has 320KB LDS;
//   we use ~167KB => 1 workgroup of 8 wave32s per WGP)

#define KDIM   172      // had_K is KDIM x KDIM
#define MDIM   64       // FWHT length / columns per batch
#define NBATCH 8192     // 2*4096
#define HS     172      // had_K LDS row stride (floats); 172 % 64 = 44 -> conflict-free A reads
#define XS     72       // X LDS row stride (floats);     72 % 64 = 8  -> conflict-free B reads
#define HROWS  176      // ceil(172/16)*16 rows allocated (rows 172..175 read-but-discarded)
#define NTILES_I 11     // ceil(172/16)
#define NTILES_N 4      // 64/16
#define LDS_FLOATS (HROWS*HS + KDIM*XS)

typedef __attribute__((ext_vector_type(2))) float v2f;
typedef __attribute__((ext_vector_type(8))) float v8f;

__global__ void __launch_bounds__(256, 1)
hadamard_wmma_kernel(const float* __restrict__ x,
                     const float* __restrict__ hadK,
                     float* __restrict__ out)
{
    extern __shared__ float lds[];
    float* hs = lds;                 // [HROWS * HS]  had_K image (stride == row length -> flat copy)
    float* xs = lds + HROWS * HS;    // [KDIM * XS]   FWHT(X)*scale, padded stride

    const int tid  = threadIdx.x;
    const int lane = tid & 31;
    const int wv   = tid >> 5;       // wave id 0..7
    const int lo   = lane & 15;      // lane group position
    const int hi   = lane >> 4;      // 0: lanes 0-15, 1: lanes 16-31

    const float scale = 0.0095311607f; // 1/sqrt(11008)

    // ---- load had_K into LDS once (HS == 172 == row length -> flat contiguous copy) ----
    for (int e = tid; e < KDIM * KDIM; e += 256) hs[e] = hadK[e];
    __syncthreads();

    for (int b = blockIdx.x; b < NBATCH; b += gridDim.x) {
        // ---------------- Phase 1: per-row FWHT_64 in registers ----------------
        if (tid < KDIM) {
            float v[MDIM];
            const float4* src = (const float4*)(x + ((size_t)b * KDIM + tid) * MDIM);
#pragma unroll
            for (int i = 0; i < MDIM / 4; ++i) {
                float4 t = src[i];
                v[4*i+0] = t.x; v[4*i+1] = t.y; v[4*i+2] = t.z; v[4*i+3] = t.w;
            }
            // 6 butterfly stages, Sylvester order (matches reference fwht)
#pragma unroll
            for (int h = 1; h < MDIM; h <<= 1) {
#pragma unroll
                for (int i = 0; i < MDIM; i += 2 * h) {
#pragma unroll
                    for (int k2 = 0; k2 < h; ++k2) {
                        float a = v[i + k2], c = v[i + k2 + h];
                        v[i + k2]     = a + c;
                        v[i + k2 + h] = a - c;
                    }
                }
            }
            float* dst = xs + tid * XS;
#pragma unroll
            for (int i = 0; i < MDIM / 4; ++i) {
                float4 t;
                t.x = v[4*i+0] * scale; t.y = v[4*i+1] * scale;
                t.z = v[4*i+2] * scale; t.w = v[4*i+3] * scale;
                *(float4*)(dst + 4 * i) = t;
            }
        }

        // prefetch next batch's X into cache while we do the matmul
        {
            int bn = b + gridDim.x;
            if (bn < NBATCH) {
                const char* p = (const char*)(x + (size_t)bn * KDIM * MDIM) + tid * 172;
                __builtin_prefetch(p, 0, 3);
            }
        }
        __syncthreads();

        // ---------------- Phase 2: out = had_K @ Y via V_WMMA_F32_16X16X4_F32 ----------------
        // 44 output tiles of 16x16, distributed over 8 waves; K-loop = 43 steps of 4.
        for (int t = wv; t < NTILES_I * NTILES_N; t += 8) {
            const int ti = t >> 2;         // row tile   (0..10)
            const int tn = t & 3;          // col tile   (0..3)
            const int arow = ti * 16 + lo; // A fragment row (may be >=172 for ti=10: discarded)
            const int ncol = tn * 16 + lo; // B/D fragment column

            v8f acc = {};
            for (int j = 0; j < KDIM; j += 4) {
                // A (16x4): lanes 0-15 hold K=j,j+1; lanes 16-31 hold K=j+2,j+3 (8B contiguous)
                const float* ap = hs + arow * HS + j + 2 * hi;
                v2f a; a.x = ap[0]; a.y = ap[1];
                // B (4x16): mirrored layout — lane group by N, VGPR pair by K
                v2f bb;
                bb.x = xs[(j + 2 * hi)     * XS + ncol];
                bb.y = xs[(j + 2 * hi + 1) * XS + ncol];
                acc = __builtin_amdgcn_wmma_f32_16x16x4_f32(
                    /*neg_a=*/false, a, /*neg_b=*/false, bb,
                    /*c_mod=*/(short)0, acc, /*reuse_a=*/false, /*reuse_b=*/false);
            }

            // D (16x16 f32): VGPR v holds rows M=v (lanes 0-15) and M=v+8 (lanes 16-31)
#pragma unroll
            for (int vj = 0; vj < 8; ++vj) {
                const int i = ti * 16 + vj + 8 * hi;
                if (i < KDIM)
                    out[((size_t)b * KDIM + i) * MDIM + ncol] = acc[vj];
            }
        }
        __syncthreads(); // xs reused next batch
    }
}

extern "C" void kernel_launch(void* const* d_in, const int* in_sizes, int n_in,
                              void* d_out, int out_size, void* d_ws, size_t ws_size,
                              hipStream_t stream)
{
    (void)in_sizes; (void)n_in; (void)out_size; (void)d_ws; (void)ws_size;
    const float* x    = (const float*)d_in[0];
    const float* hadK = (const float*)d_in[1];
    float*       out  = (float*)d_out;

    dim3 grid(1024), block(256);
    size_t shmem = (size_t)LDS_FLOATS * sizeof(float); // ~167 KB of the WGP's 320 KB LDS
    hipLaunchKernelGGL(hadamard_wmma_kernel, grid, block, shmem, stream, x, hadK, out);
}